// TO_BlockNet_61770219651542
// MI455X (gfx1250) — compile-verified
//
#include <hip/hip_runtime.h>
#include <math.h>

// Problem constants from the reference
#define N_ROWS   4
#define N_COLS   4
#define N_DEPTHS 2
#define N_BLOCKS 32   // 4*4*2
#define N_BASIS  50
#define M_PAD    64   // basis padded to 4 WMMA tiles of 16
#define W_STRIDE 65   // 65 mod 64 == 1 -> bank = (block + m) mod 64, conflict-free

typedef __attribute__((ext_vector_type(2))) float v2f;
typedef __attribute__((ext_vector_type(8))) float v8f;

// ---- monotone float<->u32 key (total order incl. negatives) ----
__device__ __forceinline__ unsigned enc_f32(float f) {
    unsigned u = __float_as_uint(f);
    return (u & 0x80000000u) ? ~u : (u | 0x80000000u);
}
__device__ __forceinline__ float dec_f32(unsigned k) {
    unsigned u = (k & 0x80000000u) ? (k & 0x7fffffffu) : ~k;
    return __uint_as_float(u);
}

// ws[0..2] = min keys (y,x,z), ws[3..5] = max keys. Re-init every call:
// the harness poisons ws once and never restores it between replays.
__global__ void TO_BlockNet_init_ws(unsigned* __restrict__ mm) {
    int t = threadIdx.x;
    if (t < 3)      mm[t] = 0xFFFFFFFFu; // +inf key
    else if (t < 6) mm[t] = 0x00000000u; // -inf key
}

// Pass 1: global coordinate min/max. Wave butterfly then 6 u32 atomics/wave.
__global__ __launch_bounds__(256)
void TO_BlockNet_minmax(const float* __restrict__ coords, int n,
                        unsigned* __restrict__ mm) {
    int i = blockIdx.x * blockDim.x + threadIdx.x;
    int q = i < n ? i : n - 1; // duplicates don't change min/max
    float y = coords[3 * q + 0], x = coords[3 * q + 1], z = coords[3 * q + 2];
    unsigned ky = enc_f32(y), kx = enc_f32(x), kz = enc_f32(z);
    unsigned miny = ky, minx = kx, minz = kz, maxy = ky, maxx = kx, maxz = kz;
#pragma unroll
    for (int o = 1; o < 32; o <<= 1) {
        miny = min(miny, (unsigned)__shfl_xor((int)miny, o, 32));
        minx = min(minx, (unsigned)__shfl_xor((int)minx, o, 32));
        minz = min(minz, (unsigned)__shfl_xor((int)minz, o, 32));
        maxy = max(maxy, (unsigned)__shfl_xor((int)maxy, o, 32));
        maxx = max(maxx, (unsigned)__shfl_xor((int)maxx, o, 32));
        maxz = max(maxz, (unsigned)__shfl_xor((int)maxz, o, 32));
    }
    if ((threadIdx.x & 31) == 0) {
        atomicMin(&mm[0], miny); atomicMin(&mm[1], minx); atomicMin(&mm[2], minz);
        atomicMax(&mm[3], maxy); atomicMax(&mm[4], maxx); atomicMax(&mm[5], maxz);
    }
}

// Pass 2: fused gather-first evaluation.
// One wave owns a 16-point tile. z-tile(16 pts x 16 basis) comes from
// V_WMMA_F32_16X16X4_F32 with A = [y x z 1] (16x4) and B = [ky;kx;kz;1] (4x16).
// Basis padded 50->64 with zero K columns / zero weights in LDS so every
// ds_load is unconditional (no exec-mask branching around LDS fetches).
__global__ __launch_bounds__(256)
void TO_BlockNet_main(const float* __restrict__ coords,
                      const float* __restrict__ kernels,  // (32,3,50), broadcast of one 3x50 K
                      const float* __restrict__ weights,  // (32,50)
                      const float* __restrict__ bias,     // (32,)
                      const unsigned* __restrict__ mm,
                      float* __restrict__ out, int n) {
    __shared__ float k_pad[3 * M_PAD];              // kernels[0], zero-padded cols 50..63
    __shared__ float w_pad[N_BLOCKS * W_STRIDE];    // weights, stride 65, zero-padded
    __shared__ float b_lds[N_BLOCKS];

    int tid = threadIdx.x;
    for (int i = tid; i < 3 * M_PAD; i += 256) {
        int row = i >> 6, col = i & 63;
        k_pad[i] = (col < N_BASIS) ? kernels[row * N_BASIS + col] : 0.0f;
    }
    for (int i = tid; i < N_BLOCKS * W_STRIDE; i += 256) {
        int row = i / W_STRIDE, col = i - row * W_STRIDE;
        w_pad[i] = (col < N_BASIS) ? weights[row * N_BASIS + col] : 0.0f;
    }
    if (tid < N_BLOCKS) b_lds[tid] = bias[tid];
    __syncthreads();

    // decode global min/max
    float my = dec_f32(mm[0]), mx = dec_f32(mm[1]), mz = dec_f32(mm[2]);
    float My = dec_f32(mm[3]), Mx = dec_f32(mm[4]), Mz = dec_f32(mm[5]);

    int lane = tid & 31;
    int wave = tid >> 5;
    int p0   = (blockIdx.x * 8 + wave) * 16;     // 16-point tile per wave
    int q    = p0 + (lane & 15);
    int qc   = q < n ? q : n - 1;                // EXEC stays all-ones for WMMA

    float cy = coords[3 * qc + 0], cx = coords[3 * qc + 1], cz = coords[3 * qc + 2];

    // block index, matching reference float math: (d / (M-m+1e-9)) * dims, trunc, clamp
    float ty = (cy - my) / (My - my + 1e-9f) * (float)N_ROWS;
    float tx = (cx - mx) / (Mx - mx + 1e-9f) * (float)N_COLS;
    float tz = (cz - mz) / (Mz - mz + 1e-9f) * (float)N_DEPTHS;
    int r = min(max((int)ty, 0), N_ROWS - 1);
    int c = min(max((int)tx, 0), N_COLS - 1);
    int d = min(max((int)tz, 0), N_DEPTHS - 1);
    int bq = r * (N_COLS * N_DEPTHS) + c * N_DEPTHS + d;   // lane (l&15): point's block

    bool hi = lane >= 16;
    // A layout (32-bit A 16x4): v0 = K0(lanes0-15)/K2(lanes16-31); v1 = K1/K3
    v2f A;
    A[0] = hi ? cz : cy;     // K=0 -> y, K=2 -> z
    A[1] = hi ? 1.0f : cx;   // K=1 -> x, K=3 -> ones column (the "+1")

    // broadcast tile block ids: VGPR v holds points v (lanes<16) / v+8 (lanes>=16)
    int wbase[8];
#pragma unroll
    for (int v = 0; v < 8; ++v)
        wbase[v] = __shfl(bq, v + (hi ? 8 : 0), 32) * W_STRIDE;

    float acc[8] = {0.f, 0.f, 0.f, 0.f, 0.f, 0.f, 0.f, 0.f};

#pragma unroll
    for (int ch = 0; ch < 4; ++ch) {                 // 50 basis -> 4 tiles of 16
        int m = ch * 16 + (lane & 15);
        // unconditional fetch from zero-padded LDS; padded cols: z=1, weight=0
        float kr0 = k_pad[m];                        // ky[m]
        float kr1 = k_pad[M_PAD + m];                // kx[m]
        float kr2 = k_pad[2 * M_PAD + m];            // kz[m]
        v2f B;
        B[0] = hi ? kr2 : kr0;                       // rows K=0 / K=2
        B[1] = hi ? 1.0f : kr1;                      // rows K=1 / K=3 (ones row)
        v8f C = {};
        // z-tile = coords_tile @ [K;1]  (16x16 f32 accumulator, 8 VGPRs)
        v8f Z = __builtin_amdgcn_wmma_f32_16x16x4_f32(
            false, A, false, B, (short)0, C, false, false);
#pragma unroll
        for (int v = 0; v < 8; ++v) {
            float sv = __sinf(Z[v]);                 // native v_sin_f32, |z| < 64
            acc[v] += sv * w_pad[wbase[v] + m];      // conflict-free: bank=(b+m)%64
        }
    }

    // reduce 16 basis lanes within each half; lane0 -> point v, lane16 -> point v+8
#pragma unroll
    for (int v = 0; v < 8; ++v) {
        float s = acc[v];
        s += __shfl_xor(s, 1, 32);
        s += __shfl_xor(s, 2, 32);
        s += __shfl_xor(s, 4, 32);
        s += __shfl_xor(s, 8, 32);
        if ((lane & 15) == 0) {
            int p = p0 + v + (hi ? 8 : 0);
            if (p < n) {
                float val = s + b_lds[wbase[v] / W_STRIDE];
                out[p] = 1.0f / (1.0f + __expf(-val));     // sigmoid
            }
        }
    }
}

extern "C" void kernel_launch(void* const* d_in, const int* in_sizes, int n_in,
                              void* d_out, int out_size, void* d_ws, size_t ws_size,
                              hipStream_t stream) {
    const float* coords  = (const float*)d_in[0];
    const float* kernels = (const float*)d_in[1];
    const float* weights = (const float*)d_in[2];
    const float* bias    = (const float*)d_in[3];
    float*       out     = (float*)d_out;
    unsigned*    mm      = (unsigned*)d_ws;   // 6 u32 keys

    int n = in_sizes[0] / 3;                  // 131072

    TO_BlockNet_init_ws<<<1, 64, 0, stream>>>(mm);
    TO_BlockNet_minmax<<<(n + 255) / 256, 256, 0, stream>>>(coords, n, mm);
    // 256 threads = 8 waves, 16 points/wave -> 128 points/block
    TO_BlockNet_main<<<(n + 127) / 128, 256, 0, stream>>>(coords, kernels, weights,
                                                          bias, mm, out, n);
}